// GraphConvolution_71923522339430
// MI455X (gfx1250) — compile-verified
//
#include <hip/hip_runtime.h>

typedef __attribute__((ext_vector_type(2))) float v2f;
typedef __attribute__((ext_vector_type(8))) float v8f;

#define N_NODES 50000
#define N_EDGES 800000
#define D_IN    128
#define D_OUT   64
#define N_STRIPS (N_NODES / 16)   // 3125 exactly
#define KPSTRIDE 160              // floats per K-pair row in LDS; 160%64==32 -> bank-conflict-free

// ---------------------------------------------------------------------------
// Stage 1: h[N,64] = x[N,128] @ W[128,64] using V_WMMA_F32_16X16X4_F32.
// 256 threads/block = 8 wave32; each wave computes one 16-row strip x 64 cols
// as four 16x16 f32 accumulator tiles, looping K in steps of 4.
//
// A (16x4 f32) layout: lanes 0-15 -> M=lane, VGPRs = K0/K1; lanes 16-31 -> K2/K3.
// B (4x16): lane%16 = N, lane/16 selects K-pair.
// W is staged in LDS K-pair interleaved: (k,c) -> [(k>>1)*KPSTRIDE + 2c + (k&1)]
// so each B fragment is ONE contiguous, aligned ds_load_b64 (no reg shuffles),
// and the two half-wave K-pairs land in disjoint bank halves (stride 160).
// ---------------------------------------------------------------------------
__global__ __launch_bounds__(256) void gcn_gemm_wmma(const float* __restrict__ x,
                                                     const float* __restrict__ W,
                                                     float* __restrict__ h) {
  __shared__ float lw[(D_IN / 2) * KPSTRIDE];   // 64 * 160 * 4B = 40 KB
  for (int i = threadIdx.x; i < D_IN * D_OUT; i += 256) {
    const int k = i >> 6;          // 0..127
    const int c = i & 63;          // 0..63
    lw[(k >> 1) * KPSTRIDE + 2 * c + (k & 1)] = W[i];
  }
  __syncthreads();

  const int wave  = threadIdx.x >> 5;
  const int lane  = threadIdx.x & 31;
  const int strip = blockIdx.x * 8 + wave;
  if (strip >= N_STRIPS) return;

  const int r0    = strip * 16;
  const int m     = lane & 15;                  // row-in-tile (A) / col-in-tile (B, C/D)
  const int khalf = lane >> 4;                  // selects K pair {0,1} vs {2,3}

  v8f acc0 = {}, acc1 = {}, acc2 = {}, acc3 = {};

  const float* xrow = x + (size_t)(r0 + m) * D_IN;

  for (int kb = 0; kb < D_IN; kb += 4) {
    const int k0 = kb + khalf * 2;
    v2f a;
    a.x = xrow[k0];
    a.y = xrow[k0 + 1];

    // B fragments: contiguous v2f loads from interleaved LDS
    const float* wp = &lw[((kb >> 1) + khalf) * KPSTRIDE + 2 * m];
    const v2f b0 = *(const v2f*)(wp);
    const v2f b1 = *(const v2f*)(wp + 32);
    const v2f b2 = *(const v2f*)(wp + 64);
    const v2f b3 = *(const v2f*)(wp + 96);

    acc0 = __builtin_amdgcn_wmma_f32_16x16x4_f32(false, a, false, b0, (short)0, acc0, false, false);
    acc1 = __builtin_amdgcn_wmma_f32_16x16x4_f32(false, a, false, b1, (short)0, acc1, false, false);
    acc2 = __builtin_amdgcn_wmma_f32_16x16x4_f32(false, a, false, b2, (short)0, acc2, false, false);
    acc3 = __builtin_amdgcn_wmma_f32_16x16x4_f32(false, a, false, b3, (short)0, acc3, false, false);
  }

  // Write back: VGPR r holds row (khalf*8 + r), column lane%16 of each tile.
  float* hp = h + (size_t)r0 * D_OUT;
  const int rbase = khalf * 8;
#pragma unroll
  for (int r = 0; r < 8; ++r) {
    const int row = rbase + r;
    hp[row * D_OUT + m]      = acc0[r];
    hp[row * D_OUT + m + 16] = acc1[r];
    hp[row * D_OUT + m + 32] = acc2[r];
    hp[row * D_OUT + m + 48] = acc3[r];
  }
}

// ---------------------------------------------------------------------------
// Stage 2: out[i,:] = bias + sum_{e : dst[e]==i} val[e] * h[src[e],:]
// edge_dst is SORTED -> each node owns a contiguous edge range; find it by
// binary search, accumulate in registers, single store. No atomics, fully
// deterministic, coalesced 256B reads of h per edge (L2-resident).
// Team = 64 threads (one per channel); 4 teams per 256-thread block.
// ---------------------------------------------------------------------------
__global__ __launch_bounds__(256) void gcn_aggregate(const float* __restrict__ h,
                                                     const int* __restrict__ esrc,
                                                     const int* __restrict__ edst,
                                                     const float* __restrict__ eval,
                                                     const float* __restrict__ bias,
                                                     float* __restrict__ out) {
  const int c    = threadIdx.x;                       // channel 0..63
  const int node = blockIdx.x * 4 + threadIdx.y;
  if (node >= N_NODES) return;

  // lower_bound(edst, node)
  int lo = 0, hi = N_EDGES;
  while (lo < hi) {
    int mid = (lo + hi) >> 1;
    if (edst[mid] < node) lo = mid + 1; else hi = mid;
  }
  const int e0 = lo;
  // upper_bound(edst, node)
  hi = N_EDGES;
  while (lo < hi) {
    int mid = (lo + hi) >> 1;
    if (edst[mid] <= node) lo = mid + 1; else hi = mid;
  }
  const int e1 = lo;

  float acc = 0.0f;
  for (int e = e0; e < e1; ++e) {
    const int   s = esrc[e];                          // wave-uniform -> scalar load
    const float v = eval[e];
    acc += v * h[(size_t)s * D_OUT + c];
  }
  out[(size_t)node * D_OUT + c] = acc + bias[c];
}

extern "C" void kernel_launch(void* const* d_in, const int* in_sizes, int n_in,
                              void* d_out, int out_size, void* d_ws, size_t ws_size,
                              hipStream_t stream) {
  const float* x    = (const float*)d_in[0];
  const int*   esrc = (const int*)d_in[1];
  const int*   edst = (const int*)d_in[2];
  const float* eval = (const float*)d_in[3];
  const float* W    = (const float*)d_in[4];
  const float* bias = (const float*)d_in[5];
  float*       out  = (float*)d_out;
  float*       h    = (float*)d_ws;                   // 50000*64*4 = 12.8 MB scratch

  gcn_gemm_wmma<<<(N_STRIPS + 7) / 8, 256, 0, stream>>>(x, W, h);

  dim3 blk(64, 4);
  gcn_aggregate<<<(N_NODES + 3) / 4, blk, 0, stream>>>(h, esrc, edst, eval, bias, out);
}